// NLPGERM_11269994185355
// MI455X (gfx1250) — compile-verified
//
#include <hip/hip_runtime.h>
#include <math.h>

#define NCLS  5
#define FEAT  2048
#define DD    1024
#define BB    64
#define EPSV  0.05f
#define NITER 50

typedef float v2f __attribute__((ext_vector_type(2)));
typedef float v8f __attribute__((ext_vector_type(8)));

// ---------------------------------------------------------------------------
// Kernel 1: Z[c][b][d] = sum_f fea[b][f] * W[c][d][f]
// One wave per (class, 16-wide N tile); computes all 4 M tiles (M=64) so the
// B fragment is loaded once per K-step and reused by 4 WMMAs.
// V_WMMA_F32_16X16X4_F32, fp32 end to end (matches reference precision).
// ---------------------------------------------------------------------------
__global__ __launch_bounds__(32) void gemm_wmma(const float* __restrict__ fea,
                                                const float* __restrict__ W,
                                                float* __restrict__ Z) {
    const int c    = blockIdx.y;
    const int n0   = blockIdx.x * 16;
    const int lane = threadIdx.x;      // single wave32
    const int half = lane >> 4;        // 0: K=k,k+1   1: K=k+2,k+3
    const int r    = lane & 15;

    v8f acc[4];
#pragma unroll
    for (int m = 0; m < 4; ++m)
#pragma unroll
        for (int q = 0; q < 8; ++q) acc[m][q] = 0.0f;

    // B fragment source: B[kk][n] = W[c][n0+n][kk]
    const float* brow = W + ((size_t)c * DD + (size_t)(n0 + r)) * FEAT + 2 * half;
    // A fragment sources: A[m][kk] = fea[mtile*16 + m][kk]
    const float* arow[4];
#pragma unroll
    for (int m = 0; m < 4; ++m)
        arow[m] = fea + (size_t)(m * 16 + r) * FEAT + 2 * half;

    for (int k = 0; k < FEAT; k += 4) {
        v2f bf = *(const v2f*)(brow + k);   // 8B aligned: k%4==0, +2*half
#pragma unroll
        for (int m = 0; m < 4; ++m) {
            v2f af = *(const v2f*)(arow[m] + k);
            acc[m] = __builtin_amdgcn_wmma_f32_16x16x4_f32(
                false, af, false, bf, (short)0, acc[m], false, false);
        }
    }

    // C/D layout: VGPR rr -> M=rr (lanes 0-15) / M=rr+8 (lanes 16-31); N = lane&15
#pragma unroll
    for (int m = 0; m < 4; ++m)
#pragma unroll
        for (int rr = 0; rr < 8; ++rr) {
            int row = m * 16 + rr + 8 * half;
            int col = n0 + r;
            Z[((size_t)c * BB + row) * DD + col] = acc[m][rr];
        }
}

// ---------------------------------------------------------------------------
// Kernel 2: gather per-sample class, add bias, log-softmax (row in LDS)
// loga[b] = logsoftmax(Z[y[b]][b] + bW[y[b]]), logb[b] = logsoftmax(anchor[y[b]])
// ---------------------------------------------------------------------------
__device__ __forceinline__ void logsoftmax_block(const float* __restrict__ buf,
                                                 float* __restrict__ out,
                                                 float* red, int t) {
    float m = -__builtin_inff();
    for (int i = t; i < DD; i += 256) m = fmaxf(m, buf[i]);
    red[t] = m; __syncthreads();
    for (int s = 128; s > 0; s >>= 1) {
        if (t < s) red[t] = fmaxf(red[t], red[t + s]);
        __syncthreads();
    }
    m = red[0]; __syncthreads();
    float sum = 0.f;
    for (int i = t; i < DD; i += 256) sum += __expf(buf[i] - m);
    red[t] = sum; __syncthreads();
    for (int s = 128; s > 0; s >>= 1) {
        if (t < s) red[t] += red[t + s];
        __syncthreads();
    }
    float ls = m + __logf(red[0]); __syncthreads();
    for (int i = t; i < DD; i += 256) out[i] = buf[i] - ls;
}

__global__ __launch_bounds__(256) void logsm_kernel(const float* __restrict__ Z,
                                                    const float* __restrict__ bW,
                                                    const float* __restrict__ anchor,
                                                    const int* __restrict__ y,
                                                    float* __restrict__ loga,
                                                    float* __restrict__ logb) {
    const int b = blockIdx.x;
    const int t = threadIdx.x;
    __shared__ float red[256];
    __shared__ float mbuf[DD];
    __shared__ float abuf[DD];
    const int cls = y[b];
    for (int i = t; i < DD; i += 256) {
        mbuf[i] = Z[((size_t)cls * BB + b) * DD + i] + bW[(size_t)cls * DD + i];
        abuf[i] = anchor[(size_t)cls * DD + i];
    }
    __syncthreads();
    logsoftmax_block(mbuf, loga + (size_t)b * DD, red, t);
    __syncthreads();
    logsoftmax_block(abuf, logb + (size_t)b * DD, red, t);
}

// ---------------------------------------------------------------------------
// Kernel 3: full Sinkhorn for one (batch, pair) per block; all state in LDS.
// pair 0: (a,b)  pair 1: (a,a)  pair 2: (b,b). Cost M_ij=(i-j)^2 in-register.
// Thread i owns row/col index i; j-sweeps are lockstep LDS broadcasts.
// ---------------------------------------------------------------------------
__global__ __launch_bounds__(1024) void sinkhorn_kernel(const float* __restrict__ loga_all,
                                                        const float* __restrict__ logb_all,
                                                        float* __restrict__ ot) {
    const int b = blockIdx.x;
    const int p = blockIdx.y;
    const int i = threadIdx.x;
    __shared__ float sLr[DD];   // log of first marginal  (rows, f)
    __shared__ float sLc[DD];   // log of second marginal (cols, g)
    __shared__ float sU[DD];    // scratch potentials / reduction
    __shared__ float sG[DD];

    const float* rl = (p == 2 ? logb_all : loga_all) + (size_t)b * DD;
    const float* cl = (p == 1 ? loga_all : logb_all) + (size_t)b * DD;
    sLr[i] = rl[i];
    sLc[i] = cl[i];

    const float inv_eps = 1.0f / EPSV;
    const float fif = (float)i;
    float fi = 0.f, gi = 0.f;

    for (int it = 0; it < NITER; ++it) {
        __syncthreads();                       // protect sU overwrite
        sU[i] = gi * inv_eps + sLc[i];         // u_j = g_j/eps + log b_j
        __syncthreads();
        float m = -__builtin_inff();
        for (int j = 0; j < DD; ++j) {
            float d = fif - (float)j;
            m = fmaxf(m, sU[j] - d * d * inv_eps);
        }
        float s = 0.f;
        for (int j = 0; j < DD; ++j) {
            float d = fif - (float)j;
            s += __expf(sU[j] - d * d * inv_eps - m);
        }
        fi = -EPSV * (m + __logf(s));          // new f uses old g

        __syncthreads();                       // all reads of sU done
        sU[i] = fi * inv_eps + sLr[i];         // u_i = f_i/eps + log a_i
        __syncthreads();
        m = -__builtin_inff();
        for (int j = 0; j < DD; ++j) {
            float d = (float)j - fif;
            m = fmaxf(m, sU[j] - d * d * inv_eps);
        }
        s = 0.f;
        for (int j = 0; j < DD; ++j) {
            float d = (float)j - fif;
            s += __expf(sU[j] - d * d * inv_eps - m);
        }
        gi = -EPSV * (m + __logf(s));          // new g uses new f
    }

    __syncthreads();
    sG[i] = gi;
    __syncthreads();

    // value = sum_ij exp((f_i+g_j - (i-j)^2)/eps + loga_i + logb_j) * (f_i+g_j)
    float part = 0.f;
    const float li = sLr[i];
    for (int j = 0; j < DD; ++j) {
        float fg = fi + sG[j];
        float d  = fif - (float)j;
        part += __expf((fg - d * d) * inv_eps + li + sLc[j]) * fg;
    }
    __syncthreads();
    sU[i] = part;
    __syncthreads();
    for (int s2 = 512; s2 > 0; s2 >>= 1) {
        if (i < s2) sU[i] += sU[i + s2];
        __syncthreads();
    }
    if (i == 0) ot[p * BB + b] = sU[0];
}

// ---------------------------------------------------------------------------
// Kernel 4: losses = 2*ot_ab - ot_aa - ot_bb; weights = softmax(-losses);
// loss = sum(losses*weights). out[0]=loss, out[1..64]=weights.
// ---------------------------------------------------------------------------
__global__ __launch_bounds__(64) void final_kernel(const float* __restrict__ ot,
                                                   float* __restrict__ out) {
    const int t = threadIdx.x;   // 64 threads
    __shared__ float red[64];
    float l  = 2.f * ot[t] - ot[BB + t] - ot[2 * BB + t];
    float nl = -l;
    red[t] = nl; __syncthreads();
    for (int s = 32; s > 0; s >>= 1) {
        if (t < s) red[t] = fmaxf(red[t], red[t + s]);
        __syncthreads();
    }
    float m = red[0]; __syncthreads();
    float w = __expf(nl - m);
    red[t] = w; __syncthreads();
    for (int s = 32; s > 0; s >>= 1) {
        if (t < s) red[t] += red[t + s];
        __syncthreads();
    }
    float sum = red[0]; __syncthreads();
    w /= sum;
    out[1 + t] = w;
    red[t] = l * w; __syncthreads();
    for (int s = 32; s > 0; s >>= 1) {
        if (t < s) red[t] += red[t + s];
        __syncthreads();
    }
    if (t == 0) out[0] = red[0];
}

// ---------------------------------------------------------------------------
extern "C" void kernel_launch(void* const* d_in, const int* in_sizes, int n_in,
                              void* d_out, int out_size, void* d_ws, size_t ws_size,
                              hipStream_t stream) {
    const float* fea    = (const float*)d_in[0];   // [64,2048]
    const int*   y      = (const int*)  d_in[1];   // [64]
    const float* W      = (const float*)d_in[2];   // [5,1024,2048]
    const float* bW     = (const float*)d_in[3];   // [5,1024]
    const float* anchor = (const float*)d_in[4];   // [5,1024]
    float*       out    = (float*)d_out;           // [65]

    float* ws   = (float*)d_ws;
    float* Z    = ws;                               // 5*64*1024
    float* loga = Z    + (size_t)NCLS * BB * DD;    // 64*1024
    float* logb = loga + (size_t)BB * DD;           // 64*1024
    float* ot   = logb + (size_t)BB * DD;           // 3*64

    gemm_wmma     <<<dim3(DD / 16, NCLS), 32,   0, stream>>>(fea, W, Z);
    logsm_kernel  <<<BB,                  256,  0, stream>>>(Z, bW, anchor, y, loga, logb);
    sinkhorn_kernel<<<dim3(BB, 3),        1024, 0, stream>>>(loga, logb, ot);
    final_kernel  <<<1,                   64,   0, stream>>>(ot, out);
}